// ExperienceGuidedRetrieval_9423158248249
// MI455X (gfx1250) — compile-verified
//
#include <hip/hip_runtime.h>
#include <hip/hip_bf16.h>

typedef float v8f __attribute__((ext_vector_type(8)));
typedef float v4f __attribute__((ext_vector_type(4)));
typedef float v2f __attribute__((ext_vector_type(2)));
typedef int   v4i __attribute__((ext_vector_type(4)));

#define DIM 1024
#define TILE_M 16
#define KCHUNK 64
#define A_STRIDE (KCHUNK + 4)   /* 68 dwords: LDS banks = (4*row + col) % 64, conflict-free; 272B row pitch keeps 16B align */
#define CTX_STRIDE (DIM + 4)    /* 1028 dwords: same bank property */
#define WPB 4                   /* waves per 128-thread block */

#if defined(__gfx1250__) && __has_builtin(__builtin_amdgcn_global_load_async_to_lds_b128) && \
    __has_builtin(__builtin_amdgcn_s_wait_asynccnt)
#define EGR_ASYNC 1
typedef __attribute__((address_space(1))) v4i as1_v4i;
typedef __attribute__((address_space(3))) v4i as3_v4i;
#else
#define EGR_ASYNC 0
#endif

// ---------------------------------------------------------------------------
// Tiny MLP GEMM: Y[b][j] = act( sum_k X[b][k] * W[k][j] + bias[j] ), B <= 4
// ---------------------------------------------------------------------------
__global__ void egr_mlp_gemm(const float* __restrict__ X,
                             const float* __restrict__ W,
                             const float* __restrict__ bias,
                             float* __restrict__ Y, int B, int relu) {
    __shared__ float xs[4 * DIM];
    const int tid = threadIdx.x;
    for (int i = tid; i < 4 * DIM; i += blockDim.x)
        xs[i] = (i < B * DIM) ? X[i] : 0.0f;
    __syncthreads();

    const int j = blockIdx.x * blockDim.x + tid;
    float acc[4] = {0.f, 0.f, 0.f, 0.f};
    for (int k = 0; k < DIM; ++k) {
        const float w = W[k * DIM + j];
#pragma unroll
        for (int b = 0; b < 4; ++b) acc[b] += xs[b * DIM + k] * w;
    }
    const float bj = bias[j];
    for (int b = 0; b < B; ++b) {
        float v = acc[b] + bj;
        if (relu) v = fmaxf(v, 0.0f);
        Y[b * DIM + j] = v;
    }
}

// ---------------------------------------------------------------------------
// LayerNorm + cosine-normalize one context row per block (256 threads).
// ---------------------------------------------------------------------------
__device__ __forceinline__ float block_sum(float x, float* red) {
    const int tid = threadIdx.x;
    red[tid] = x;
    __syncthreads();
    for (int off = 128; off > 0; off >>= 1) {
        if (tid < off) red[tid] += red[tid + off];
        __syncthreads();
    }
    const float r = red[0];
    __syncthreads();
    return r;
}

__global__ void egr_ln_norm(const float* __restrict__ Yin,
                            const float* __restrict__ gamma,
                            const float* __restrict__ beta,
                            float* __restrict__ ctxn) {
    __shared__ float red[256];
    const int b = blockIdx.x, tid = threadIdx.x;
    const float* y = Yin + b * DIM;

    float v[4];
#pragma unroll
    for (int i = 0; i < 4; ++i) v[i] = y[tid + i * 256];

    const float mu = block_sum(v[0] + v[1] + v[2] + v[3], red) * (1.0f / DIM);

    float q = 0.f;
#pragma unroll
    for (int i = 0; i < 4; ++i) { const float d = v[i] - mu; q += d * d; }
    const float var = block_sum(q, red) * (1.0f / DIM);
    const float rstd = rsqrtf(var + 1e-5f);

    float z[4];
    float q2 = 0.f;
#pragma unroll
    for (int i = 0; i < 4; ++i) {
        z[i] = (v[i] - mu) * rstd * gamma[tid + i * 256] + beta[tid + i * 256];
        q2 += z[i] * z[i];
    }
    const float nrm = sqrtf(block_sum(q2, red));
    const float inv = 1.0f / fmaxf(nrm, 1e-8f);
#pragma unroll
    for (int i = 0; i < 4; ++i) ctxn[b * DIM + tid + i * 256] = z[i] * inv;
}

// ---------------------------------------------------------------------------
// Stage one 16-row x 64-col f32 tile into LDS.
// B128: one instruction stages two rows (lanes 0-15 -> row r, 16-31 -> r+1).
// Rows clamped (not skipped) so exactly 8 async ops issue per stage,
// keeping s_wait_asynccnt constants exact. Clamped rows are never stored.
// ---------------------------------------------------------------------------
__device__ __forceinline__ void egr_stage(float* __restrict__ dst,
                                          const float* __restrict__ ep,
                                          int base, int kc, int lane, int N) {
    const int half = lane >> 4;
    const int l16  = lane & 15;
#pragma unroll
    for (int r = 0; r < TILE_M; r += 2) {
        int row = base + r + half;
        row = (row < N) ? row : (N - 1);
        const float* gsrc = ep + (size_t)row * DIM + kc + 4 * l16;
        float*       lds  = dst + (r + half) * A_STRIDE + 4 * l16;
#if EGR_ASYNC
        __builtin_amdgcn_global_load_async_to_lds_b128(
            (as1_v4i*)gsrc, (as3_v4i*)lds, 0, 0);
#else
        *(v4f*)lds = *(const v4f*)gsrc;
#endif
    }
}

__device__ __forceinline__ void egr_wait_async(int pending) {
#if EGR_ASYNC
    if (pending) __builtin_amdgcn_s_wait_asynccnt(8);
    else         __builtin_amdgcn_s_wait_asynccnt(0);
#else
    (void)pending;
#endif
}

// ---------------------------------------------------------------------------
// Main streaming kernel. One wave owns a 16-episode tile:
//   D[16x16] = A(16 eps x 4) * B(4 x 16 ctx cols), K=1024 via 256 WMMAs.
// Double-buffered async global->LDS staging overlaps the WMMA consume.
// ep-row sum-of-squares accumulated for free from A fragments.
// ---------------------------------------------------------------------------
__global__ void __launch_bounds__(128) egr_score_wmma(
    const float* __restrict__ ep, const float* __restrict__ elapsed,
    const float* __restrict__ ctxn, float* __restrict__ out,
    int N, int B, int ntiles) {
    __shared__ float ctx_lds[5 * CTX_STRIDE];                    // rows 0..B-1 data, row B zeros
    __shared__ float a_lds[WPB * 2 * TILE_M * A_STRIDE];         // double buffer per wave

    const int tid = threadIdx.x;
    for (int i = tid; i < 5 * CTX_STRIDE; i += blockDim.x) {
        const int r = i / CTX_STRIDE, cc = i % CTX_STRIDE;
        ctx_lds[i] = (r < B && cc < DIM) ? ctxn[r * DIM + cc] : 0.0f;
    }
    __syncthreads();

    const int lane = tid & 31;
    const int wave = tid >> 5;
    const int n    = lane & 15;   // batch column (and A-matrix row) for this lane
    const int half = lane >> 4;
    const int brow = (n < B ? n : B) * CTX_STRIDE;               // zero row for n >= B
    float* aT0 = a_lds + wave * 2 * TILE_M * A_STRIDE;
    float* aT1 = aT0 + TILE_M * A_STRIDE;

    const int gwaves = gridDim.x * WPB;
    for (int tile = blockIdx.x * WPB + wave; tile < ntiles; tile += gwaves) {
        const int base = tile * TILE_M;
        v8f acc = {};
        float ssq = 0.0f;

        egr_stage(aT0, ep, base, 0, lane, N);                    // prologue: chunk 0 in flight

        for (int kc = 0; kc < DIM; kc += KCHUNK) {
            float* cur = ((kc / KCHUNK) & 1) ? aT1 : aT0;
            float* nxt = ((kc / KCHUNK) & 1) ? aT0 : aT1;
            const int havenext = (kc + KCHUNK < DIM);
            if (havenext) egr_stage(nxt, ep, base, kc + KCHUNK, lane, N);
            egr_wait_async(havenext);                            // current chunk landed

            // 16 WMMA steps over this K chunk
#pragma unroll
            for (int ks = 0; ks < KCHUNK; ks += 4) {
                const v2f a = *(const v2f*)(cur + n * A_STRIDE + ks + half * 2);
                ssq += a.x * a.x + a.y * a.y;
                const v2f bf = *(const v2f*)(ctx_lds + brow + kc + ks + half * 2);
                acc = __builtin_amdgcn_wmma_f32_16x16x4_f32(
                    /*neg_a=*/false, a, /*neg_b=*/false, bf,
                    /*c_mod=*/(short)0, acc, /*reuse_a=*/false, /*reuse_b=*/false);
            }
        }

        // Per-episode-row ||ep||^2: lanes l and l+16 hold the two halves.
        const float ssq_full = ssq + __shfl_xor(ssq, 16);
        const int rowl = base + n;
        const float el = (rowl < N) ? elapsed[rowl] : 0.0f;
        const float w = expf(-0.1f * el) + 0.1f;                 // boost*score = 0.2*0.5
        const float fac = w / fmaxf(sqrtf(ssq_full), 1e-8f);     // lane n holds row base+n

        // D layout: VGPR v, lanes 0-15 -> M=v, lanes 16-31 -> M=v+8; N = lane%16
#pragma unroll
        for (int v = 0; v < 8; ++v) {
            const int mrow = v + 8 * half;
            const float fv = __shfl(fac, mrow);
            float sc = acc[v] * fv;
            sc = (sc > 0.7f) ? sc : 0.0f;
            const int grow = base + mrow;
            if (n < B && grow < N) out[(size_t)n * N + grow] = sc;
        }
    }
}

// ---------------------------------------------------------------------------
extern "C" void kernel_launch(void* const* d_in, const int* in_sizes, int n_in,
                              void* d_out, int out_size, void* d_ws, size_t ws_size,
                              hipStream_t stream) {
    const float* ctx     = (const float*)d_in[0];   // [B, D]
    const float* ep      = (const float*)d_in[1];   // [N, D]
    const float* elapsed = (const float*)d_in[2];   // [N]
    const float* W1      = (const float*)d_in[3];
    const float* b1      = (const float*)d_in[4];
    const float* W2      = (const float*)d_in[5];
    const float* b2      = (const float*)d_in[6];
    const float* g       = (const float*)d_in[7];
    const float* be      = (const float*)d_in[8];

    int B = in_sizes[0] / DIM;
    if (B > 4) B = 4;
    const int N = in_sizes[2];

    float* ws = (float*)d_ws;
    float* H  = ws;                 // [4, D]
    float* Y  = ws + 4 * DIM;       // [4, D]
    float* Cn = ws + 8 * DIM;       // [4, D]

    egr_mlp_gemm<<<DIM / 128, 128, 0, stream>>>(ctx, W1, b1, H, B, 1);
    egr_mlp_gemm<<<DIM / 128, 128, 0, stream>>>(H, W2, b2, Y, B, 0);
    egr_ln_norm<<<B, 256, 0, stream>>>(Y, g, be, Cn);

    const int ntiles = (N + TILE_M - 1) / TILE_M;
    int nblocks = (ntiles + WPB - 1) / WPB;
    if (nblocks > 32768) nblocks = 32768;
    egr_score_wmma<<<nblocks, 128, 0, stream>>>(ep, elapsed, Cn, (float*)d_out,
                                                N, B, ntiles);
}